// Model_33921651704514
// MI455X (gfx1250) — compile-verified
//
#include <hip/hip_runtime.h>

// ---------------------------------------------------------------------------
// Mamba-2 SSD chunked scan for MI455X (gfx1250, wave32, WMMA).
// Memory-bound problem (~15 GFLOP vs ~1 GB traffic); all GEMM work is pushed
// through V_WMMA_F32_16X16X4_F32 so VALU stays free and global traffic is
// b128-coalesced. FP32 throughout (reference dtype).
// ---------------------------------------------------------------------------

typedef __attribute__((ext_vector_type(2))) float v2f;
typedef __attribute__((ext_vector_type(8))) float v8f;

#define BB   8
#define LL   8192
#define HH   16
#define PP   64
#define NN   16
#define CLEN 64
#define NC   (LL / CLEN)   // 128 chunks

#define XPAD 72            // LDS row stride for 64-wide rows (bank-conflict pad)
#define BPAD 20            // LDS row stride for 16-wide rows

static __device__ __forceinline__ v8f wmma4(v2f a, v2f b, v8f c) {
  // D(16x16,f32) = A(16x4,f32) * B(4x16,f32) + C
  return __builtin_amdgcn_wmma_f32_16x16x4_f32(false, a, false, b,
                                               (short)0, c, false, false);
}

// inclusive Hillis-Steele scan of a_cum[0..63] in LDS (all threads barrier)
static __device__ __forceinline__ void scan64(float* a_cum, int tid) {
  for (int off = 1; off < CLEN; off <<= 1) {
    float v = 0.0f;
    if (tid < CLEN && tid >= off) v = a_cum[tid - off];
    __syncthreads();
    if (tid < CLEN) a_cum[tid] += v;
    __syncthreads();
  }
}

// ---------------------------------------------------------------------------
// Kernel 1: per-(b,c,h) intra-chunk state  states[p][n] = sum_l X[l][p]*B[l][n]*decay[l]
// 128 threads = 4 waves; wave w computes p-rows [16w,16w+16) via 16 chained WMMAs.
// ---------------------------------------------------------------------------
__global__ __launch_bounds__(128)
void k_states_intra(const float* __restrict__ X, const float* __restrict__ A,
                    const float* __restrict__ Bm, float* __restrict__ states,
                    float* __restrict__ csum) {
  __shared__ float a_cum[CLEN];
  __shared__ float Bs[CLEN * BPAD];
  __shared__ float Xs[CLEN * XPAD];

  const int bid = blockIdx.x;
  const int h = bid % HH;
  const int c = (bid / HH) % NC;
  const int b = bid / (HH * NC);
  const int tid = threadIdx.x;
  const int lane = tid & 31;
  const int w = tid >> 5;

  const long rowX0 = ((long)(b * LL + c * CLEN) * HH + h) * PP;
  const long rowB0 = ((long)(b * LL + c * CLEN) * HH + h) * NN;
  const long rowA0 = (long)(b * LL + c * CLEN) * HH + h;

  if (tid < CLEN) a_cum[tid] = A[rowA0 + (long)tid * HH];
#pragma unroll
  for (int i = 0; i < 8; ++i) {              // X chunk: 64x64 = 1024 float4
    int idx = tid + i * 128, l = idx >> 4, q = idx & 15;
    *(float4*)(&Xs[l * XPAD + q * 4]) =
        *(const float4*)(X + rowX0 + (long)l * (HH * PP) + q * 4);
  }
#pragma unroll
  for (int i = 0; i < 2; ++i) {              // B chunk: 64x16 = 256 float4
    int idx = tid + i * 128, l = idx >> 2, q = idx & 3;
    *(float4*)(&Bs[l * BPAD + q * 4]) =
        *(const float4*)(Bm + rowB0 + (long)l * (HH * NN) + q * 4);
  }
  __syncthreads();
  scan64(a_cum, tid);
  const float a_last = a_cum[CLEN - 1];
#pragma unroll
  for (int i = 0; i < 8; ++i) {              // scale B rows by decay
    int idx = tid + i * 128, l = idx >> 4, n = idx & 15;
    Bs[l * BPAD + n] *= __expf(a_last - a_cum[l]);
  }
  __syncthreads();

  const int colL = lane & 15;
  const int khL  = (lane >> 4) << 1;
  v8f acc = {};
#pragma unroll
  for (int kk = 0; kk < 16; ++kk) {          // K = 64 (l dimension)
    int k = kk * 4 + khL;
    const float* pa = &Xs[k * XPAD + w * 16 + colL];   // A[m=p][k=l] = X[l][p]
    v2f af; af[0] = pa[0]; af[1] = pa[XPAD];
    const float* pb = &Bs[k * BPAD + colL];            // B[k=l][col=n]
    v2f bf; bf[0] = pb[0]; bf[1] = pb[BPAD];
    acc = wmma4(af, bf, acc);
  }
  const long sbase = ((long)((b * HH + h) * NC + c)) * (PP * NN);
#pragma unroll
  for (int v = 0; v < 8; ++v) {
    int p = w * 16 + v + ((lane >> 4) << 3);
    states[sbase + p * NN + colL] = acc[v];
  }
  if (tid == 0) csum[(b * HH + h) * NC + c] = a_last;
}

// ---------------------------------------------------------------------------
// Kernel 2: inter-chunk recurrence. Elementwise per (p,n): 1024 independent
// scans per (b,h). Writes the *entering* state back in place (reference inter).
// ---------------------------------------------------------------------------
__global__ __launch_bounds__(256)
void k_scan(const float* __restrict__ init, float* __restrict__ states,
            const float* __restrict__ csum) {
  const int bid = blockIdx.x;
  const int split = bid & 3;
  const int bh = bid >> 2;
  const int e = split * 256 + threadIdx.x;   // 0..1023 = p*16+n
  float s = init[(long)bh * (PP * NN) + e];
  const long base = (long)bh * NC * (PP * NN) + e;
  const float* cs = csum + bh * NC;
  for (int c = 0; c < NC; ++c) {
    long idx = base + (long)c * (PP * NN);
    float intra = states[idx];
    float dec = __expf(cs[c]);
    states[idx] = s;                         // entering state for chunk c
    s = dec * s + intra;
  }
}

// ---------------------------------------------------------------------------
// Kernel 3: per-(b,c,h) output  Y = (maskdecay(C·Bᵀ))·X + diag(e^{ac})·C·stateᵀ
// 4 waves; wave w owns l-rows [16w,16w+16): builds masked G strip (16x64) in
// LDS, then 4 output tiles with K=64 + K=16 chained WMMAs.
// ---------------------------------------------------------------------------
__global__ __launch_bounds__(128)
void k_output(const float* __restrict__ X, const float* __restrict__ A,
              const float* __restrict__ Bm, const float* __restrict__ Cm,
              const float* __restrict__ states, float* __restrict__ Y) {
  __shared__ float a_cum[CLEN];
  __shared__ float Cs[CLEN * BPAD];
  __shared__ float Bs[CLEN * BPAD];
  __shared__ float Ss[CLEN * BPAD];          // state [p][n]
  __shared__ float Xs[CLEN * XPAD];
  __shared__ float Gs[4 * 16 * XPAD];        // per-wave masked G strips

  const int bid = blockIdx.x;
  const int h = bid % HH;
  const int c = (bid / HH) % NC;
  const int b = bid / (HH * NC);
  const int tid = threadIdx.x;
  const int lane = tid & 31;
  const int w = tid >> 5;

  const long rowX0 = ((long)(b * LL + c * CLEN) * HH + h) * PP;
  const long rowB0 = ((long)(b * LL + c * CLEN) * HH + h) * NN;
  const long rowA0 = (long)(b * LL + c * CLEN) * HH + h;
  const long sbase = ((long)((b * HH + h) * NC + c)) * (PP * NN);

  if (tid < CLEN) a_cum[tid] = A[rowA0 + (long)tid * HH];
#pragma unroll
  for (int i = 0; i < 8; ++i) {
    int idx = tid + i * 128, l = idx >> 4, q = idx & 15;
    *(float4*)(&Xs[l * XPAD + q * 4]) =
        *(const float4*)(X + rowX0 + (long)l * (HH * PP) + q * 4);
  }
#pragma unroll
  for (int i = 0; i < 2; ++i) {
    int idx = tid + i * 128, l = idx >> 2, q = idx & 3;
    *(float4*)(&Bs[l * BPAD + q * 4]) =
        *(const float4*)(Bm + rowB0 + (long)l * (HH * NN) + q * 4);
    *(float4*)(&Cs[l * BPAD + q * 4]) =
        *(const float4*)(Cm + rowB0 + (long)l * (HH * NN) + q * 4);
    *(float4*)(&Ss[l * BPAD + q * 4]) =
        *(const float4*)(states + sbase + l * NN + q * 4);
  }
  __syncthreads();
  scan64(a_cum, tid);

  const int colL = lane & 15;
  const int khL  = (lane >> 4) << 1;
  const int tm = w;
  float al[8];
#pragma unroll
  for (int v = 0; v < 8; ++v) al[v] = a_cum[tm * 16 + v + ((lane >> 4) << 3)];
  const float el = __expf(a_cum[tm * 16 + colL]);   // exp(ac[l]) for A-frag rows
  float* gw = &Gs[w * 16 * XPAD];

  // --- masked G strip: G[l][s] = exp(ac[l]-ac[s]) * sum_n C[l,n]B[s,n], l>=s
#pragma unroll
  for (int sb = 0; sb < 4; ++sb) {
    v8f g = {};
#pragma unroll
    for (int kk = 0; kk < 4; ++kk) {               // K = 16 (n dimension)
      int k = kk * 4 + khL;
      const float* pa = &Cs[(tm * 16 + colL) * BPAD + k];  // A[m=l][k=n]
      v2f af; af[0] = pa[0]; af[1] = pa[1];
      const float* pb = &Bs[(sb * 16 + colL) * BPAD + k];  // B[k=n][col=s]=Bc[s][n]
      v2f bf; bf[0] = pb[0]; bf[1] = pb[1];
      g = wmma4(af, bf, g);
    }
    const int sidx = sb * 16 + colL;
    const float as = a_cum[sidx];
#pragma unroll
    for (int v = 0; v < 8; ++v) {
      int l = tm * 16 + v + ((lane >> 4) << 3);
      g[v] *= (l >= sidx) ? __expf(al[v] - as) : 0.0f;
    }
#pragma unroll
    for (int v = 0; v < 8; ++v) {
      int m = v + ((lane >> 4) << 3);
      gw[m * XPAD + sb * 16 + colL] = g[v];        // D-layout -> LDS (A-major)
    }
  }

  // --- output tiles: Y = G·X + (C∘e^{ac})·stateᵀ
  const long yrow0 = ((long)(b * LL + c * CLEN + tm * 16) * HH + h) * PP;
#pragma unroll
  for (int tn = 0; tn < 4; ++tn) {
    v8f acc = {};
#pragma unroll
    for (int kk = 0; kk < 4; ++kk) {               // Y_off, K = 16 (n)
      int k = kk * 4 + khL;
      const float* pa = &Cs[(tm * 16 + colL) * BPAD + k];
      v2f af; af[0] = pa[0] * el; af[1] = pa[1] * el;
      const float* pb = &Ss[(tn * 16 + colL) * BPAD + k];  // B[k=n][col=p]=state[p][n]
      v2f bf; bf[0] = pb[0]; bf[1] = pb[1];
      acc = wmma4(af, bf, acc);
    }
#pragma unroll
    for (int kk = 0; kk < 16; ++kk) {              // Y_diag, K = 64 (s)
      int k = kk * 4 + khL;
      const float* pa = &gw[colL * XPAD + k];      // A[m=l][k=s] from G strip
      v2f af; af[0] = pa[0]; af[1] = pa[1];
      const float* pb = &Xs[k * XPAD + tn * 16 + colL];    // B[k=s][col=p]
      v2f bf; bf[0] = pb[0]; bf[1] = pb[XPAD];
      acc = wmma4(af, bf, acc);
    }
#pragma unroll
    for (int v = 0; v < 8; ++v) {
      int m = v + ((lane >> 4) << 3);
      Y[yrow0 + (long)m * (HH * PP) + tn * 16 + colL] = acc[v];
    }
  }
}

// ---------------------------------------------------------------------------
extern "C" void kernel_launch(void* const* d_in, const int* in_sizes, int n_in,
                              void* d_out, int out_size, void* d_ws, size_t ws_size,
                              hipStream_t stream) {
  const float* X    = (const float*)d_in[0];
  const float* A    = (const float*)d_in[1];
  const float* Bm   = (const float*)d_in[2];
  const float* Cm   = (const float*)d_in[3];
  const float* init = (const float*)d_in[4];
  float* Y = (float*)d_out;

  float* states = (float*)d_ws;                               // b*h*c*p*n
  float* csum   = states + (size_t)BB * HH * NC * PP * NN;    // b*h*c

  k_states_intra<<<BB * NC * HH, 128, 0, stream>>>(X, A, Bm, states, csum);
  k_scan<<<BB * HH * 4, 256, 0, stream>>>(init, states, csum);
  k_output<<<BB * NC * HH, 128, 0, stream>>>(X, A, Bm, Cm, states, Y);
}